// RNN_13176959664222
// MI455X (gfx1250) — compile-verified
//
#include <hip/hip_runtime.h>

#define GN 4096
#define UNR 8

typedef __attribute__((ext_vector_type(2))) float v2f;
typedef __attribute__((ext_vector_type(8))) float v8f;

__device__ __forceinline__ float sigmoidf_(float x) {
    return 1.0f / (1.0f + expf(-x));
}

__device__ __forceinline__ float reluf_(float x) {
    return x > 0.0f ? x : 0.0f;
}

// Finalize one gen_index query from the wave-uniform argmin result.
// sg points at the LDS copy of g (address space recovered after inlining).
__device__ __forceinline__ float finalize_gen_index(const float* sg, float ac, int arg0) {
    int arg = arg0;
    if (sg[arg] > ac) arg += 1;                 // jnp.where(x0[arg] > ac, arg+1, arg)
    float hi = sg[(arg + GN - 1) % GN];         // x0[arg-1] with JAX negative wrap
    float lo = sg[arg];
    const float step = (float)(GN - 1) * 0.5f;  // 2047.5
    return (hi - ac) / (hi - lo) / step + (float)(arg - 1) / step - 1.0f;
}

// Batched cross-lane argmin reduction: min dist, tie -> min index.
template <int NQ>
__device__ __forceinline__ void wave_argmin_reduce(float* best, int* bidx) {
    #pragma unroll
    for (int off = 16; off > 0; off >>= 1) {
        #pragma unroll
        for (int q = 0; q < NQ; ++q) {
            float ob = __shfl_xor(best[q], off, 32);
            int   oi = __shfl_xor(bidx[q], off, 32);
            if (ob < best[q] || (ob == best[q] && oi < bidx[q])) {
                best[q] = ob; bidx[q] = oi;
            }
        }
    }
}

__device__ float bilinear_(const float* __restrict__ t, float gx, float gy) {
    const int W = GN, H = GN;
    float px = fminf(fmaxf((gx + 1.0f) * 0.5f * (float)(W - 1), 0.0f), (float)(W - 1));
    float py = fminf(fmaxf((gy + 1.0f) * 0.5f * (float)(H - 1), 0.0f), (float)(H - 1));
    float x0f = floorf(px), y0f = floorf(py);
    float wx = px - x0f, wy = py - y0f;
    int x0 = (int)x0f; x0 = x0 < 0 ? 0 : (x0 > W - 1 ? W - 1 : x0);
    int x1 = x0 + 1 > W - 1 ? W - 1 : x0 + 1;
    int y0 = (int)y0f; y0 = y0 < 0 ? 0 : (y0 > H - 1 ? H - 1 : y0);
    int y1 = y0 + 1 > H - 1 ? H - 1 : y0 + 1;
    float t00 = t[(size_t)y0 * W + x0];
    float t01 = t[(size_t)y0 * W + x1];
    float t10 = t[(size_t)y1 * W + x0];
    float t11 = t[(size_t)y1 * W + x1];
    return t00 * (1.0f - wx) * (1.0f - wy) + t01 * wx * (1.0f - wy)
         + t10 * (1.0f - wx) * wy + t11 * wx * wy;
}

__global__ void __launch_bounds__(32)
fused_gs_kernel(const float* __restrict__ x,        // (3,2)
                const float* __restrict__ s_weight, // (2,)
                const float* __restrict__ t_weight, // (3,2)
                const float* __restrict__ t_weight2,// (2,2)
                const float* __restrict__ g,        // (4096,)
                const float* __restrict__ res1,
                const float* __restrict__ res2,
                const float* __restrict__ alpha1,   // (3,)
                const float* __restrict__ alpha2,   // (3,)
                const float* __restrict__ soa_res1,
                const float* __restrict__ soa_res2,
                float* __restrict__ out)            // [y00, y01, regu]
{
    __shared__ float s_g[GN];   // 16 KB staging of g for stage-B scan + finalize taps

    const int lane = (int)threadIdx.x;
    const int i = lane < 3 ? lane : 0;  // lanes >=3 replicate point 0 (results unused)

    const float flag = x[2 * i + 0];
    const float sw0 = sigmoidf_(s_weight[0]);
    const float sw1 = sigmoidf_(s_weight[1]);
    const float gfirst = g[0];
    const float glast = g[GN - 1];

    // ---- Stage A: 5 independent gen_index queries, ONE pass over g ----
    // Manual 8x unroll: 8 clause-grouped global loads up front per outer
    // iteration (overlapped latency), mirrored into LDS for stage B.
    float acA[5] = { x[1], x[3], x[5], sw0, sw1 };
    float bestA[5];
    int bidxA[5];
    #pragma unroll
    for (int q = 0; q < 5; ++q) {
        acA[q] = fminf(fmaxf(acA[q], glast), gfirst);  // clip(a, x0[-1], x0[0])
        bestA[q] = 3.4e38f;
        bidxA[q] = 0x7fffffff;
    }
    for (int j0 = lane; j0 < GN; j0 += 32 * UNR) {
        float gv[UNR];
        #pragma unroll
        for (int u = 0; u < UNR; ++u) gv[u] = g[j0 + 32 * u];
        #pragma unroll
        for (int u = 0; u < UNR; ++u) {
            const int j = j0 + 32 * u;       // ascending per lane -> first-min kept
            s_g[j] = gv[u];
            #pragma unroll
            for (int q = 0; q < 5; ++q) {
                float d = fabsf(acA[q] - gv[u]);
                if (d < bestA[q]) { bestA[q] = d; bidxA[q] = j; }
            }
        }
    }
    wave_argmin_reduce<5>(bestA, bidxA);

    float qo[5];
    #pragma unroll
    for (int q = 0; q < 5; ++q) qo[q] = finalize_gen_index(s_g, acA[q], bidxA[q]);

    const float gx1 = qo[3];  // gen_index(sw0)
    const float gx2 = qo[4];  // gen_index(sw1)
    const float gy = (i == 0) ? qo[0] : ((i == 1) ? qo[1] : qo[2]);

    // per-point bilinear taps (lanes 0..2 carry real points)
    const float y1g = bilinear_(res1, gx1, gy);
    const float y2g = bilinear_(res2, gx2, gy);
    const float a1v = (flag == 1.0f ? y1g : 0.0f) * alpha1[i];
    const float a2v = (flag == 2.0f ? y2g : 0.0f) * alpha2[i];

    // broadcast the 3-vectors to every lane
    const float a1_0 = __shfl(a1v, 0, 32), a1_1 = __shfl(a1v, 1, 32), a1_2 = __shfl(a1v, 2, 32);
    const float a2_0 = __shfl(a2v, 0, 32), a2_1 = __shfl(a2v, 1, 32), a2_2 = __shfl(a2v, 2, 32);

    const float tw00 = sigmoidf_(t_weight[0]), tw01 = sigmoidf_(t_weight[1]);
    const float tw10 = sigmoidf_(t_weight[2]), tw11 = sigmoidf_(t_weight[3]);
    const float tw20 = sigmoidf_(t_weight[4]), tw21 = sigmoidf_(t_weight[5]);

    // ---- Both tiny GEMMs in one V_WMMA_F32_16X16X4_F32 ----
    // A (16x4 f32): lanes 0-15 hold {K0,K1}, lanes 16-31 hold {K2,K3} for row M=lane%16.
    //   Row 0 = y1 triple, Row 1 = y2 triple.
    v2f A = { 0.0f, 0.0f };
    A.x = (lane == 0) ? a1_0 : (lane == 1) ? a2_0 : (lane == 16) ? a1_2 : (lane == 17) ? a2_2 : 0.0f;
    A.y = (lane == 0) ? a1_1 : (lane == 1) ? a2_1 : 0.0f;
    // B (4x16 f32): lane n holds {B[0][n],B[1][n]}, lane 16+n holds {B[2][n],B[3][n]}.
    v2f Bm = { 0.0f, 0.0f };
    Bm.x = (lane == 0) ? tw00 : (lane == 1) ? tw01 : (lane == 16) ? tw20 : (lane == 17) ? tw21 : 0.0f;
    Bm.y = (lane == 0) ? tw10 : (lane == 1) ? tw11 : 0.0f;

    v8f C = { 0.0f, 0.0f, 0.0f, 0.0f, 0.0f, 0.0f, 0.0f, 0.0f };
    // (neg_a, A, neg_b, B, c_mod, C, reuse_a, reuse_b) — EXEC is all-1s here.
    v8f D = __builtin_amdgcn_wmma_f32_16x16x4_f32(false, A, false, Bm, (short)0, C, false, false);

    // D row M=r lives in VGPR r, N=lane (lanes 0-15)
    const float y1o0 = __shfl(D[0], 0, 32);  // y1 @ tw, col 0
    const float y1o1 = __shfl(D[0], 1, 32);  // y1 @ tw, col 1
    const float y2o0 = __shfl(D[1], 0, 32);  // y2 @ tw, col 0
    const float y2o1 = __shfl(D[1], 1, 32);  // y2 @ tw, col 1

    // ---- Stage B: 4 independent gen_index queries, ONE pass over LDS copy ----
    float acB[4] = { y1o0, y2o0, y1o1, y2o1 };
    float bestB[4];
    int bidxB[4];
    #pragma unroll
    for (int q = 0; q < 4; ++q) {
        acB[q] = fminf(fmaxf(acB[q], glast), gfirst);
        bestB[q] = 3.4e38f;
        bidxB[q] = 0x7fffffff;
    }
    for (int j0 = lane; j0 < GN; j0 += 32 * UNR) {
        float gv[UNR];
        #pragma unroll
        for (int u = 0; u < UNR; ++u) gv[u] = s_g[j0 + 32 * u];  // ds loads, no bank conflicts
        #pragma unroll
        for (int u = 0; u < UNR; ++u) {
            const int j = j0 + 32 * u;
            #pragma unroll
            for (int q = 0; q < 4; ++q) {
                float d = fabsf(acB[q] - gv[u]);
                if (d < bestB[q]) { bestB[q] = d; bidxB[q] = j; }
            }
        }
    }
    wave_argmin_reduce<4>(bestB, bidxB);

    float qbo[4];
    #pragma unroll
    for (int q = 0; q < 4; ++q) qbo[q] = finalize_gen_index(s_g, acB[q], bidxB[q]);
    const float gy_1 = qbo[0], gx_1 = qbo[1];
    const float gy_2 = qbo[2], gx_2 = qbo[3];

    const float y_1 = bilinear_(soa_res1, gx_1, gy_1);
    const float y_2 = bilinear_(soa_res2, gx_2, gy_2);

    // n==1 so mean() == identity;  (... )/mean(a)/2
    const float regu1 = (reluf_(y1o0 - gfirst - 0.001f) + reluf_(glast + 0.001f - y1o0)) / y1o0 * 0.5f
                      + (reluf_(y2o0 - gfirst - 0.001f) + reluf_(glast + 0.001f - y2o0)) / y2o0 * 0.5f;
    const float regu2 = (reluf_(y1o1 - gfirst - 0.001f) + reluf_(glast + 0.001f - y1o1)) / y1o1 * 0.5f
                      + (reluf_(y2o1 - gfirst - 0.001f) + reluf_(glast + 0.001f - y2o1)) / y2o1 * 0.5f;

    const float t200 = sigmoidf_(t_weight2[0]), t201 = sigmoidf_(t_weight2[1]);
    const float t210 = sigmoidf_(t_weight2[2]), t211 = sigmoidf_(t_weight2[3]);

    if (lane == 0) {
        out[0] = y_1 * t200 + y_2 * t210;
        out[1] = y_1 * t201 + y_2 * t211;
        out[2] = regu1 + regu2;
    }
}

extern "C" void kernel_launch(void* const* d_in, const int* in_sizes, int n_in,
                              void* d_out, int out_size, void* d_ws, size_t ws_size,
                              hipStream_t stream) {
    (void)in_sizes; (void)n_in; (void)out_size; (void)d_ws; (void)ws_size;
    const float* x         = (const float*)d_in[0];
    const float* s_weight  = (const float*)d_in[1];
    const float* t_weight  = (const float*)d_in[2];
    const float* t_weight2 = (const float*)d_in[3];
    const float* g         = (const float*)d_in[4];
    const float* res1      = (const float*)d_in[5];
    const float* res2      = (const float*)d_in[6];
    const float* alpha1    = (const float*)d_in[7];
    const float* alpha2    = (const float*)d_in[8];
    const float* soa_res1  = (const float*)d_in[9];
    const float* soa_res2  = (const float*)d_in[10];
    float* out = (float*)d_out;

    fused_gs_kernel<<<1, 32, 0, stream>>>(x, s_weight, t_weight, t_weight2, g,
                                          res1, res2, alpha1, alpha2,
                                          soa_res1, soa_res2, out);
}